// MultiHeadGraphAttention_74612171866565
// MI455X (gfx1250) — compile-verified
//
#include <hip/hip_runtime.h>
#include <math.h>

#define B_ 8
#define CIN 64
#define T_ 10
#define M_ 325
#define NHIS 12
#define DOUT 64
#define NHEADS 8
#define DK 8
#define MM (M_*M_)           /* 105625 */
#define BTM (B_*T_*M_)       /* 26000  */
#define NEG_INF (-__builtin_inff())

typedef __attribute__((ext_vector_type(2))) float v2f;
typedef __attribute__((ext_vector_type(8))) float v8f;

#if defined(__has_builtin)
#if __has_builtin(__builtin_amdgcn_global_load_async_to_lds_b32)
#define HAS_ASYNC_LDS 1
#endif
#endif

// ---------------------------------------------------------------------------
// Kernel 1: stream distance/shortpath/movement/quater ONCE, fold channel
// reductions + 1x1 conv (N_HIS -> T) into distT/spT/mqT workspace planes.
// ---------------------------------------------------------------------------
__global__ void prep_kernel(const float* __restrict__ dist,
                            const float* __restrict__ sp,
                            const float* __restrict__ qt,
                            const float* __restrict__ mv,
                            const float* __restrict__ conv_w,
                            const float* __restrict__ conv_b,
                            const float* __restrict__ mov_w,
                            const float* __restrict__ mov_b,
                            const float* __restrict__ qua_w,
                            const float* __restrict__ qua_b,
                            float* __restrict__ distT,
                            float* __restrict__ spT,
                            float* __restrict__ mqT) {
    __shared__ float sCW[T_*NHIS];
    __shared__ float sCB[T_];
    __shared__ float sMW[3], sQW[4], sMB, sQB;
    int tid = threadIdx.x;
    if (tid < T_*NHIS) sCW[tid] = conv_w[tid];
    if (tid < T_)      sCB[tid] = conv_b[tid];
    if (tid < 3)       sMW[tid] = mov_w[tid];
    if (tid < 4)       sQW[tid] = qua_w[tid];
    if (tid == 0) { sMB = mov_b[0]; sQB = qua_b[0]; }
    __syncthreads();

    long gid = (long)blockIdx.x * blockDim.x + tid;
    if (gid >= (long)B_ * MM) return;
    int j = (int)(gid % M_);
    int i = (int)((gid / M_) % M_);
    int b = (int)(gid / MM);

    float aD[T_] = {};
    float aS[T_] = {};
    float aQ[T_] = {};
    long base2 = (long)b*NHIS*MM + (long)i*M_ + j;
    for (int n = 0; n < NHIS; ++n) {
        long idx = base2 + (long)n*MM;
        float d = dist[idx];
        float s = sp[idx];
        const float* qp = qt + idx*4;
        const float* mp = mv + idx*3;
        if (n + 1 < NHIS) {
            __builtin_prefetch(qp + (long)MM*4, 0, 1);
            __builtin_prefetch(mp + (long)MM*3, 0, 1);
        }
        float4 q4 = *reinterpret_cast<const float4*>(qp);
        float mq = mp[0]*sMW[0] + mp[1]*sMW[1] + mp[2]*sMW[2] + sMB
                 + q4.x*sQW[0] + q4.y*sQW[1] + q4.z*sQW[2] + q4.w*sQW[3] + sQB;
        #pragma unroll
        for (int t = 0; t < T_; ++t) {
            float w = sCW[t*NHIS + n];
            aD[t] = fmaf(w, d,  aD[t]);
            aS[t] = fmaf(w, s,  aS[t]);
            aQ[t] = fmaf(w, mq, aQ[t]);
        }
    }
    long obase = (long)b*T_*MM + (long)i*M_ + j;
    #pragma unroll
    for (int t = 0; t < T_; ++t) {
        long o = obase + (long)t*MM;
        distT[o] = aD[t] + sCB[t];
        spT[o]   = aS[t] + sCB[t];
        mqT[o]   = aQ[t] + 2.0f*sCB[t];   // mv and qt each carry one conv_b
    }
}

// ---------------------------------------------------------------------------
// Kernel 2: h = X * W  via V_WMMA_F32_16X16X4_F32.
// One wave computes a 16x16 tile of h (rows = flattened b,t,m ; cols = D).
// ---------------------------------------------------------------------------
__global__ void hgemm_kernel(const float* __restrict__ input,
                             const float* __restrict__ W,
                             float* __restrict__ h) {
    int wave = threadIdx.x >> 5;
    int lane = threadIdx.x & 31;
    int rowBase = blockIdx.x * 16;        // BTM = 26000 = 1625*16 exactly
    int colBase = wave * 16;
    int lr = lane & 15;
    int hi = lane >> 4;                   // 0: K+0/1, 1: K+2/3 (ISA A layout)
    int row = rowBase + lr;
    int b = row / (T_*M_);
    int rem = row % (T_*M_);
    int t = rem / M_;
    int m = rem % M_;
    long abase = (long)b*CIN*T_*M_ + (long)t*M_ + m;   // + k*T_*M_
    int col = colBase + lr;

    v8f acc = {};
    for (int k0 = 0; k0 < CIN; k0 += 4) {
        int k = k0 + 2*hi;
        v2f av, bv;
        av[0] = input[abase + (long)k     * (T_*M_)];
        av[1] = input[abase + (long)(k+1) * (T_*M_)];
        bv[0] = W[(long)k     * DOUT + col];
        bv[1] = W[(long)(k+1) * DOUT + col];
        acc = __builtin_amdgcn_wmma_f32_16x16x4_f32(false, av, false, bv,
                                                    (short)0, acc, false, false);
    }
    #pragma unroll
    for (int v = 0; v < 8; ++v) {
        int r = rowBase + v + 8*hi;       // ISA C/D layout
        h[(long)r*DOUT + col] = acc[v];
    }
}

// ---------------------------------------------------------------------------
// Kernel 3: e_i[b,h,t,m] = h . a1 ; e_j = h . a2 (rank-1, tiny)
// ---------------------------------------------------------------------------
__global__ void eij_kernel(const float* __restrict__ h,
                           const float* __restrict__ a,
                           float* __restrict__ ei,
                           float* __restrict__ ej) {
    int gid = blockIdx.x*blockDim.x + threadIdx.x;
    if (gid >= BTM*NHEADS) return;
    int head = gid % NHEADS;
    int row  = gid / NHEADS;
    int b = row / (T_*M_);
    int rem = row % (T_*M_);
    int t = rem / M_;
    int m = rem % M_;
    const float* hp = h + (long)row*DOUT + head*DK;
    float s1 = 0.f, s2 = 0.f;
    #pragma unroll
    for (int k = 0; k < DK; ++k) {
        s1 = fmaf(hp[k], a[k],      s1);
        s2 = fmaf(hp[k], a[DK + k], s2);
    }
    long o = (((long)b*NHEADS + head)*T_ + t)*M_ + m;
    ei[o] = s1;
    ej[o] = s2;
}

// ---------------------------------------------------------------------------
// Kernel 4: fused logits + softmax + (attn @ h) with WMMA accumulation.
// Block = 256 threads = 8 waves; wave w handles head w of the same
// (b, t, 16-row i tile). Logit planes staged once in LDS via the CDNA5
// async global->LDS path (no VGPR round trip), shared by all 8 heads.
// ---------------------------------------------------------------------------
__global__ void attn_kernel(const float* __restrict__ distT,
                            const float* __restrict__ spT,
                            const float* __restrict__ mqT,
                            const float* __restrict__ ei,
                            const float* __restrict__ ej,
                            const float* __restrict__ h,
                            float* __restrict__ out) {
    extern __shared__ float smem[];
    float* distS = smem;                    // 16*325
    float* spS   = distS + 16*M_;           // 16*325
    float* mqS   = spS   + 16*M_;           // 16*325
    float* ejS   = mqS   + 16*M_;           // 8*325
    float* eiS   = ejS   + NHEADS*M_;       // 8*16

    int i0 = blockIdx.x * 16;
    int bt = blockIdx.y;                    // 0..79
    int b  = bt / T_;
    int t  = bt % T_;
    int tid = threadIdx.x;

    long planeBase = ((long)bt*M_ + i0)*M_;

#ifdef HAS_ASYNC_LDS
    // The 16x325 tile of each plane is a contiguous 5200-float region in
    // global memory; copy it straight into LDS with GLOBAL_LOAD_ASYNC_TO_LDS
    // (ASYNCcnt-tracked, no VGPR staging). Edge-tile overreads stay inside
    // the workspace allocation and feed only rows masked out downstream.
    {
        const float* gsrc0 = distT + planeBase;
        const float* gsrc1 = spT   + planeBase;
        const float* gsrc2 = mqT   + planeBase;
        #define AS1P(p) ((__attribute__((address_space(1))) int*)(unsigned long long)(p))
        #define AS3P(p) ((__attribute__((address_space(3))) int*)(unsigned int)(unsigned long long)(p))
        for (int idx = tid; idx < 16*M_; idx += blockDim.x) {
            __builtin_amdgcn_global_load_async_to_lds_b32(AS1P(gsrc0 + idx), AS3P(distS + idx), 0, 0);
            __builtin_amdgcn_global_load_async_to_lds_b32(AS1P(gsrc1 + idx), AS3P(spS   + idx), 0, 0);
            __builtin_amdgcn_global_load_async_to_lds_b32(AS1P(gsrc2 + idx), AS3P(mqS   + idx), 0, 0);
        }
        #undef AS1P
        #undef AS3P
#if defined(__has_builtin) && __has_builtin(__builtin_amdgcn_s_wait_asynccnt)
        __builtin_amdgcn_s_wait_asynccnt(0);
#else
        asm volatile("s_wait_asynccnt 0x0" ::: "memory");
#endif
    }
#else
    for (int idx = tid; idx < 16*M_; idx += blockDim.x) {
        int r = idx / M_;
        int c = idx % M_;
        bool ok = (i0 + r) < M_;
        long g = planeBase + (long)r*M_ + c;
        distS[idx] = ok ? distT[g] : 0.f;
        spS[idx]   = ok ? spT[g]   : 0.f;
        mqS[idx]   = ok ? mqT[g]   : 0.f;
    }
#endif
    for (int idx = tid; idx < NHEADS*M_; idx += blockDim.x) {
        int hh = idx / M_;
        int j  = idx % M_;
        ejS[idx] = ej[(((long)b*NHEADS + hh)*T_ + t)*M_ + j];
    }
    for (int idx = tid; idx < NHEADS*16; idx += blockDim.x) {
        int hh = idx / 16;
        int r  = idx % 16;
        bool ok = (i0 + r) < M_;
        eiS[idx] = ok ? ei[(((long)b*NHEADS + hh)*T_ + t)*M_ + i0 + r] : 0.f;
    }
    __syncthreads();

    int head = tid >> 5;
    int lane = tid & 31;

    // ---- per-row softmax stats (lanes parallel over j, rows sequential) ----
    float myM = 0.f, myInvL = 0.f;
    for (int i = 0; i < 16; ++i) {
        float eiv = eiS[head*16 + i];
        float mx = NEG_INF;
        for (int j0 = 0; j0 < M_; j0 += 32) {
            int j = j0 + lane;
            float e = NEG_INF;
            if (j < M_) {
                float t1 = (eiv + ejS[head*M_ + j] + mqS[i*M_ + j]) * distS[i*M_ + j];
                if (t1 == 0.f) t1 = NEG_INF;
                float t2 = 0.5f*t1 + 0.5f*spS[i*M_ + j];
                if (t2 == 0.f) t2 = NEG_INF;
                e = t2;
            }
            mx = fmaxf(mx, e);
        }
        for (int off = 16; off > 0; off >>= 1)
            mx = fmaxf(mx, __shfl_xor(mx, off, 32));
        float sum = 0.f;
        for (int j0 = 0; j0 < M_; j0 += 32) {
            int j = j0 + lane;
            if (j < M_) {
                float t1 = (eiv + ejS[head*M_ + j] + mqS[i*M_ + j]) * distS[i*M_ + j];
                if (t1 == 0.f) t1 = NEG_INF;
                float t2 = 0.5f*t1 + 0.5f*spS[i*M_ + j];
                if (t2 == 0.f) t2 = NEG_INF;
                sum += __expf(t2 - mx);
            }
        }
        for (int off = 16; off > 0; off >>= 1)
            sum += __shfl_xor(sum, off, 32);
        if ((lane & 15) == i) { myM = mx; myInvL = 1.0f / sum; }
    }

    // ---- WMMA pass: P computed directly in A-operand lane layout ----
    int iR = lane & 15;                    // A row / B column index
    int hi = lane >> 4;                    // K sub-pair select
    bool rowOk = (i0 + iR) < M_;
    float eiv = eiS[head*16 + iR];
    long hbase = (long)bt*M_*DOUT + (long)head*DK;

    v8f acc = {};
    for (int j0 = 0; j0 < M_; j0 += 4) {
        v2f av, bv;
        #pragma unroll
        for (int u = 0; u < 2; ++u) {
            int j = j0 + 2*hi + u;
            float p = 0.f;
            if (rowOk && j < M_) {
                float t1 = (eiv + ejS[head*M_ + j] + mqS[iR*M_ + j]) * distS[iR*M_ + j];
                if (t1 == 0.f) t1 = NEG_INF;
                float t2 = 0.5f*t1 + 0.5f*spS[iR*M_ + j];
                if (t2 == 0.f) t2 = NEG_INF;
                p = __expf(t2 - myM) * myInvL;
            }
            av[u] = p;
            bv[u] = (iR < DK && j < M_) ? h[hbase + (long)j*DOUT + iR] : 0.f;
        }
        acc = __builtin_amdgcn_wmma_f32_16x16x4_f32(false, av, false, bv,
                                                    (short)0, acc, false, false);
    }

    if (iR < DK) {                         // only N = 0..7 are real d_k cols
        #pragma unroll
        for (int v = 0; v < 8; ++v) {
            int r = i0 + v + 8*hi;
            if (r < M_)
                out[((long)bt*M_ + r)*DOUT + head*DK + iR] = acc[v];
        }
    }
}

// ---------------------------------------------------------------------------
extern "C" void kernel_launch(void* const* d_in, const int* in_sizes, int n_in,
                              void* d_out, int out_size, void* d_ws, size_t ws_size,
                              hipStream_t stream) {
    const float* input    = (const float*)d_in[0];
    const float* dist     = (const float*)d_in[1];
    const float* sp       = (const float*)d_in[2];
    const float* quater   = (const float*)d_in[3];
    const float* movement = (const float*)d_in[4];
    // d_in[5] = conv_count (unused by reference)
    const float* W        = (const float*)d_in[6];
    const float* a        = (const float*)d_in[7];
    const float* mov_w    = (const float*)d_in[8];
    const float* mov_b    = (const float*)d_in[9];
    const float* qua_w    = (const float*)d_in[10];
    const float* qua_b    = (const float*)d_in[11];
    const float* conv_w   = (const float*)d_in[12];
    const float* conv_b   = (const float*)d_in[13];
    float* out = (float*)d_out;

    float* ws    = (float*)d_ws;
    float* distT = ws;
    float* spT   = distT + (long)B_*T_*MM;
    float* mqT   = spT   + (long)B_*T_*MM;
    float* hbuf  = mqT   + (long)B_*T_*MM;
    float* eibuf = hbuf  + (long)BTM*DOUT;
    float* ejbuf = eibuf + (long)NHEADS*B_*T_*M_;

    int prepBlocks = (int)(((long)B_*MM + 255) / 256);
    prep_kernel<<<prepBlocks, 256, 0, stream>>>(dist, sp, quater, movement,
                                                conv_w, conv_b, mov_w, mov_b,
                                                qua_w, qua_b, distT, spT, mqT);

    hgemm_kernel<<<BTM/16, 128, 0, stream>>>(input, W, hbuf);

    eij_kernel<<<(BTM*NHEADS + 255)/256, 256, 0, stream>>>(hbuf, a, eibuf, ejbuf);

    dim3 grid((M_ + 15)/16, B_*T_);
    size_t smemBytes = (size_t)(3*16*M_ + NHEADS*M_ + NHEADS*16) * sizeof(float);
    attn_kernel<<<grid, 256, smemBytes, stream>>>(distT, spT, mqT,
                                                  eibuf, ejbuf, hbuf, out);
}